// GIN_33182917328951
// MI455X (gfx1250) — compile-verified
//
#include <hip/hip_runtime.h>

#define N_NODES 50000
#define N_EDGES 800000
#define DCH 128
#define BN_EPS 1e-5f
#define LDS_STRIDE 132   // 128 + 4 pad floats: bank-conflict-free fragment loads

typedef __attribute__((ext_vector_type(2))) float v2f;
typedef __attribute__((ext_vector_type(8))) float v8f;
typedef __attribute__((ext_vector_type(4))) unsigned int u32x4;
typedef __attribute__((ext_vector_type(4))) int i32x4;
typedef __attribute__((ext_vector_type(8))) int i32x8;

__device__ __forceinline__ void atomAddF(float* p, float v) {
  __hip_atomic_fetch_add(p, v, __ATOMIC_RELAXED, __HIP_MEMORY_SCOPE_AGENT);
}

// ---------------------------------------------------------------- h = (1+eps)*x
__global__ void __launch_bounds__(256)
gin_init_scale(const float* __restrict__ X, float* __restrict__ H,
               const float* __restrict__ epsArr, int layer, int n4) {
  int i = blockIdx.x * blockDim.x + threadIdx.x;
  if (i >= n4) return;
  const float s = 1.0f + epsArr[layer];
  float4 v = ((const float4*)X)[i];
  v.x *= s; v.y *= s; v.z *= s; v.w *= s;
  ((float4*)H)[i] = v;
}

// ---------------------------------------------------- H[dst] += X[src] (1 wave/edge)
__global__ void __launch_bounds__(256)
gin_scatter_add(const float* __restrict__ X, const int* __restrict__ srcI,
                const int* __restrict__ dstI, float* __restrict__ H, int nE) {
  const int e = (blockIdx.x * blockDim.x + threadIdx.x) >> 5;
  if (e >= nE) return;
  const int lane = threadIdx.x & 31;
  const int s = srcI[e];
  const int d = dstI[e];
  const float4 v = *(const float4*)(X + (size_t)s * DCH + lane * 4);
  float* hd = H + (size_t)d * DCH + lane * 4;
  atomAddF(hd + 0, v.x);
  atomAddF(hd + 1, v.y);
  atomAddF(hd + 2, v.z);
  atomAddF(hd + 3, v.w);
}

// --------------------------- Y[64x128 tile] = H @ W + b via f32 WMMA + TDM staging
// block = 256 threads = 8 waves. Wave w owns N-tile [16w,16w+16) across 4 M-subtiles.
// Each wave TDM-DMAs its own 8x128 slab of H into LDS (pad fields give stride 132).
__global__ void __launch_bounds__(256)
gin_gemm_wmma(const float* __restrict__ H, const float* __restrict__ W,
              const float* __restrict__ bias, float* __restrict__ Y, int nRows) {
  __shared__ float sh[64 * LDS_STRIDE];
  const int tid  = threadIdx.x;
  const int lane = tid & 31;
  const int wv   = tid >> 5;
  const int m0   = blockIdx.x * 64;

  // ---- TDM stage: one tensor_load_to_lds per wave (8 rows of 128 f32) ----
  {
    const int row0 = m0 + wv * 8;
    int rem = nRows - row0;
    if (rem < 0) rem = 0;
    if (rem > 8) rem = 8;
    const float* gsrc = (rem > 0) ? (H + (size_t)row0 * DCH) : H;
    const unsigned long long ga = (unsigned long long)(uintptr_t)gsrc;
    const unsigned ldsAddr =
        ((unsigned)(uintptr_t)&sh[0]) + (unsigned)(wv * 8 * LDS_STRIDE * 4);

    u32x4 g0;
    g0[0] = 1u;                                       // count=1, user mode
    g0[1] = ldsAddr;                                  // D#.lds_addr
    g0[2] = (unsigned)(ga & 0xffffffffull);           // global_addr[31:0]
    g0[3] = (unsigned)((ga >> 32) & 0x01ffffffull)    // global_addr[56:32]
            | (2u << 30);                             // type=2 (image)
    i32x8 g1;
    g1[0] = (2 << 16)        // data_size = 4B
          | (1 << 20)        // pad_enable
          | (6 << 22)        // pad_interval: 128 DWORDs
          | (3 << 25);       // pad_amount: 4 DWORDs  -> LDS row stride 132
    g1[1] = (DCH << 16);     // tensor_dim0 = 128 (low16 in bits 63:48)
    g1[2] = (rem << 16);     // tensor_dim0 hi=0 | tensor_dim1 = remaining rows
    g1[3] = (DCH << 16);     // tensor_dim1 hi=0 | tile_dim0 = 128
    g1[4] = 8;               // tile_dim1 = 8, tile_dim2 = 0
    g1[5] = DCH;             // tensor_dim0_stride = 128 (low 32)
    g1[6] = 0;               // stride0 hi | tensor_dim1_stride low
    g1[7] = 0;
    i32x4 gz = {0, 0, 0, 0};
#if __clang_major__ >= 23
    i32x8 gz8 = {0, 0, 0, 0, 0, 0, 0, 0};
    __builtin_amdgcn_tensor_load_to_lds(g0, g1, gz, gz, gz8, 0);
#else
    __builtin_amdgcn_tensor_load_to_lds(g0, g1, gz, gz, 0);
#endif
    __builtin_amdgcn_s_wait_tensorcnt(0);
  }
  __syncthreads();

  const int n0    = wv << 4;
  const int mrow  = lane & 15;          // M (A) / N (B,C) position
  const int khalf = (lane >> 4) << 1;   // lane-half K offset {0,2}

  v8f acc0 = {}, acc1 = {}, acc2 = {}, acc3 = {};
  const float* Wp = W + n0 + mrow;

#pragma unroll 4
  for (int k4 = 0; k4 < 32; ++k4) {
    const int kb = (k4 << 2) + khalf;
    v2f b;                                        // B: 4x16, W[k][n0+mrow]
    b.x = Wp[(size_t)kb * DCH];
    b.y = Wp[(size_t)(kb + 1) * DCH];
    const int abase = mrow * LDS_STRIDE + kb;     // A: 16x4 slabs, K = kb+{0,1}
    const v2f a0 = *(const v2f*)(&sh[abase]);
    const v2f a1 = *(const v2f*)(&sh[abase + 16 * LDS_STRIDE]);
    const v2f a2 = *(const v2f*)(&sh[abase + 32 * LDS_STRIDE]);
    const v2f a3 = *(const v2f*)(&sh[abase + 48 * LDS_STRIDE]);
    acc0 = __builtin_amdgcn_wmma_f32_16x16x4_f32(false, a0, false, b, (short)0, acc0, false, false);
    acc1 = __builtin_amdgcn_wmma_f32_16x16x4_f32(false, a1, false, b, (short)0, acc1, false, false);
    acc2 = __builtin_amdgcn_wmma_f32_16x16x4_f32(false, a2, false, b, (short)0, acc2, false, false);
    acc3 = __builtin_amdgcn_wmma_f32_16x16x4_f32(false, a3, false, b, (short)0, acc3, false, false);
  }

  const int ncol = n0 + mrow;
  const float bb = bias[ncol];
  const int roff = (lane >> 4) << 3;              // +8 rows for upper lane half
  const v8f accs[4] = {acc0, acc1, acc2, acc3};
#pragma unroll
  for (int t = 0; t < 4; ++t) {
    if (m0 + t * 16 < nRows) {                    // subtiles are whole-in/whole-out
      const int rbase = m0 + t * 16 + roff;
#pragma unroll
      for (int v = 0; v < 8; ++v)
        Y[(size_t)(rbase + v) * DCH + ncol] = accs[t][v] + bb;
    }
  }
}

// ---------------------------------------------------------------- BN helpers
__global__ void gin_zero_stats(float* __restrict__ stats) {
  stats[threadIdx.x] = 0.0f;                      // 256 threads: sum + sumsq
}

__global__ void __launch_bounds__(128)
gin_col_stats(const float* __restrict__ Y, float* __restrict__ stats, int nRows) {
  const int c = threadIdx.x;                      // column 0..127
  float s = 0.0f, s2 = 0.0f;
  for (int r = blockIdx.x; r < nRows; r += gridDim.x) {
    const float v = Y[(size_t)r * DCH + c];
    s += v;
    s2 += v * v;
  }
  atomAddF(&stats[c], s);
  atomAddF(&stats[DCH + c], s2);
}

__global__ void __launch_bounds__(128)
gin_bn_finalize(const float* __restrict__ stats, const float* __restrict__ gamma,
                const float* __restrict__ beta, float* __restrict__ ss, float invN) {
  const int c = threadIdx.x;
  const float mean = stats[c] * invN;
  const float var  = stats[DCH + c] * invN - mean * mean;   // biased variance
  const float sc   = gamma[c] * rsqrtf(var + BN_EPS);
  ss[c]       = sc;
  ss[DCH + c] = beta[c] - mean * sc;
}

__global__ void __launch_bounds__(256)
gin_bn_relu(float* __restrict__ Y, const float* __restrict__ ss, int n4) {
  int i = blockIdx.x * blockDim.x + threadIdx.x;
  if (i >= n4) return;
  const int c = (i & 31) * 4;                     // 32 float4 per 128-wide row
  float4 v = ((const float4*)Y)[i];
  v.x = fmaxf(v.x * ss[c + 0] + ss[DCH + c + 0], 0.0f);
  v.y = fmaxf(v.y * ss[c + 1] + ss[DCH + c + 1], 0.0f);
  v.z = fmaxf(v.z * ss[c + 2] + ss[DCH + c + 2], 0.0f);
  v.w = fmaxf(v.w * ss[c + 3] + ss[DCH + c + 3], 0.0f);
  ((float4*)Y)[i] = v;
}

// ------------------------------------------------------------------ launcher
extern "C" void kernel_launch(void* const* d_in, const int* in_sizes, int n_in,
                              void* d_out, int out_size, void* d_ws, size_t ws_size,
                              hipStream_t stream) {
  const float* x   = (const float*)d_in[0];
  const int*   ei  = (const int*)d_in[1];
  const float* eps = (const float*)d_in[2];
  const float* W1  = (const float*)d_in[3];
  const float* b1  = (const float*)d_in[4];
  const float* g1  = (const float*)d_in[5];
  const float* be1 = (const float*)d_in[6];
  const float* W2  = (const float*)d_in[7];
  const float* b2  = (const float*)d_in[8];
  const float* g2  = (const float*)d_in[9];
  const float* be2 = (const float*)d_in[10];
  const float* g0  = (const float*)d_in[11];
  const float* b0  = (const float*)d_in[12];
  float* out = (float*)d_out;

  const int N = N_NODES, E = N_EDGES, D = DCH;
  const int n4 = N * D / 4;

  float* bufA  = (float*)d_ws;            // N*D scratch
  float* stats = bufA + (size_t)N * D;    // 2*D  (sum, sumsq)
  float* ss    = stats + 2 * D;           // 2*D  (scale, shift)

  const int* srcI = ei;
  const int* dstI = ei + E;

  const dim3 blk(256);
  const int elemGrid    = (n4 + 255) / 256;
  const int scatterGrid = (E * 32) / 256;
  const int gemmGrid    = (N + 63) / 64;

  auto bn_pass = [&](float* Y, const float* gamma, const float* beta) {
    gin_zero_stats<<<1, 256, 0, stream>>>(stats);
    gin_col_stats<<<256, 128, 0, stream>>>(Y, stats, N);
    gin_bn_finalize<<<1, 128, 0, stream>>>(stats, gamma, beta, ss, 1.0f / (float)N);
    gin_bn_relu<<<elemGrid, blk, 0, stream>>>(Y, ss, n4);
  };

  // ---- layer 0: conv ( input = x, agg buffer = bufA ) ----
  gin_init_scale<<<elemGrid, blk, 0, stream>>>(x, bufA, eps, 0, n4);
  gin_scatter_add<<<scatterGrid, blk, 0, stream>>>(x, srcI, dstI, bufA, E);
  gin_gemm_wmma<<<gemmGrid, blk, 0, stream>>>(bufA, W1, b1, out, N);
  bn_pass(out, g1, be1);
  gin_gemm_wmma<<<gemmGrid, blk, 0, stream>>>(out, W2, b2, bufA, N);
  bn_pass(bufA, g2, be2);
  // inter-layer BN+ReLU (g0, b0)
  bn_pass(bufA, g0, b0);

  // ---- layer 1: conv ( input = bufA, agg buffer = out ) ----
  gin_init_scale<<<elemGrid, blk, 0, stream>>>(bufA, out, eps, 1, n4);
  gin_scatter_add<<<scatterGrid, blk, 0, stream>>>(bufA, srcI, dstI, out, E);
  gin_gemm_wmma<<<gemmGrid, blk, 0, stream>>>(out, W1 + D * D, b1 + D, bufA, N);
  bn_pass(bufA, g1 + D, be1 + D);
  gin_gemm_wmma<<<gemmGrid, blk, 0, stream>>>(bufA, W2 + D * D, b2 + D, out, N);
  bn_pass(out, g2 + D, be2 + D);
}